// GCN_26199300505906
// MI455X (gfx1250) — compile-verified
//
#include <hip/hip_runtime.h>

// ---------------------------------------------------------------------------
// GCN link predictor for MI455X (gfx1250, wave32).
//  - Dense GEMMs via V_WMMA_F32_16X16X4_F32 (exact f32, one wave = 16x16 tile)
//  - Sym-normalized aggregation via L2-resident float atomics
//  - Decode via float4 dot products (z stays resident in 192MB L2)
// ---------------------------------------------------------------------------

typedef float v2f __attribute__((ext_vector_type(2)));
typedef float v8f __attribute__((ext_vector_type(8)));

#define NNODES 100000
#define THREADS 256

// ---- degree / normalization ------------------------------------------------

__global__ void zero_f32(float* __restrict__ p, int n) {
    int i = blockIdx.x * blockDim.x + threadIdx.x;
    if (i < n) p[i] = 0.0f;
}

__global__ void count_indeg(const int* __restrict__ col, int E, float* __restrict__ deg) {
    int e = blockIdx.x * blockDim.x + threadIdx.x;
    if (e < E) atomicAdd(&deg[col[e]], 1.0f);
}

__global__ void make_dinv(const float* __restrict__ deg, float* __restrict__ dinv, int n) {
    int i = blockIdx.x * blockDim.x + threadIdx.x;
    if (i < n) dinv[i] = rsqrtf(deg[i] + 1.0f);   // +1 = self loop; deg >= 1 always
}

// ---- dense GEMM: C[M,NC] = A[M,K] @ B[K,NC], one wave per 16x16 tile -------
// A 16x4 f32 layout: lanes 0-15 hold M=0..15; VGPR0 = K=(0|2), VGPR1 = K=(1|3)
// B 4x16 f32 layout: lanes 0-15 hold N=0..15; VGPR0 = K=(0|2), VGPR1 = K=(1|3)
// C/D 16x16 f32:     VGPR r -> row r (lanes 0-15) / row r+8 (lanes 16-31)

__global__ void gemm_wmma_f32(const float* __restrict__ A, const float* __restrict__ B,
                              float* __restrict__ C, int mTiles, int K, int NC) {
    int gtid   = blockIdx.x * blockDim.x + threadIdx.x;
    int wave   = gtid >> 5;
    int lane   = threadIdx.x & 31;
    int nTiles = NC >> 4;
    int mt     = wave / nTiles;
    int nt     = wave - mt * nTiles;
    if (mt >= mTiles) return;            // whole-wave exit: EXEC all-1s inside

    int half = lane >> 4;                // 0: K pair {0,1}, 1: K pair {2,3}
    int l15  = lane & 15;

    const float* Arow = A + (size_t)(mt * 16 + l15) * K;
    const float* Bcol = B + (nt * 16 + l15);

    v8f acc = {};
    for (int k = 0; k < K; k += 4) {
        v2f a, b;
        a.x = Arow[k + 2 * half];
        a.y = Arow[k + 2 * half + 1];
        b.x = Bcol[(size_t)(k + 2 * half)     * NC];
        b.y = Bcol[(size_t)(k + 2 * half + 1) * NC];
        // (neg_a, A, neg_b, B, c_mod, C, reuse_a, reuse_b)
        acc = __builtin_amdgcn_wmma_f32_16x16x4_f32(
            false, a, false, b, (short)0, acc, false, false);
    }

    float* Crow = C + (size_t)(mt * 16 + 8 * half) * NC + nt * 16 + l15;
#pragma unroll
    for (int r = 0; r < 8; ++r)
        Crow[(size_t)r * NC] = acc[r];
}

// ---- aggregation -----------------------------------------------------------

// dst[i,:] starts as self-loop term dinv[i]^2 * src[i,:]
__global__ void init_self(const float* __restrict__ src, const float* __restrict__ dinv,
                          float* __restrict__ dst, int n, int logF) {
    int i = blockIdx.x * blockDim.x + threadIdx.x;
    if (i < n) {
        float d = dinv[i >> logF];
        dst[i] = d * d * src[i];
    }
}

// per (edge, feature) thread; consecutive lanes = consecutive features -> coalesced
__global__ void scatter_edges(const int* __restrict__ row, const int* __restrict__ col,
                              const float* __restrict__ dinv, const float* __restrict__ src,
                              float* __restrict__ dst, int E, int logF) {
    int t = blockIdx.x * blockDim.x + threadIdx.x;
    int e = t >> logF;
    if (e >= E) return;
    int f = t & ((1 << logF) - 1);
    int r = row[e], c = col[e];
    float nrm = dinv[r] * dinv[c];
    atomicAdd(&dst[((size_t)c << logF) + f], nrm * src[((size_t)r << logF) + f]);
}

__global__ void bias_act(float* __restrict__ h, const float* __restrict__ b,
                         int n, int fmask, int relu) {
    int i = blockIdx.x * blockDim.x + threadIdx.x;
    if (i < n) {
        float v = h[i] + b[i & fmask];
        h[i] = relu ? fmaxf(v, 0.0f) : v;
    }
}

// ---- decode: logits + edge_index output ------------------------------------
// out layout (floats): [0,2E) logits | [2E,4E) edge_index row0 | [4E,6E) row1

__global__ void decode(const int* __restrict__ pos, const int* __restrict__ neg,
                       const float* __restrict__ z, float* __restrict__ out, int E) {
    int e  = blockIdx.x * blockDim.x + threadIdx.x;
    int E2 = 2 * E;
    if (e >= E2) return;
    int src, dst;
    if (e < E) { src = pos[e];     dst = pos[E + e]; }
    else       { int q = e - E; src = neg[q]; dst = neg[E + q]; }

    const float4* zs = (const float4*)(z + (size_t)src * 32);
    const float4* zd = (const float4*)(z + (size_t)dst * 32);
    float acc = 0.0f;
#pragma unroll
    for (int j = 0; j < 8; ++j) {
        float4 a = zs[j], b = zd[j];
        acc += a.x * b.x + a.y * b.y + a.z * b.z + a.w * b.w;
    }
    out[e]          = acc;
    out[E2 + e]     = (float)src;
    out[2 * E2 + e] = (float)dst;
}

// ---------------------------------------------------------------------------

extern "C" void kernel_launch(void* const* d_in, const int* in_sizes, int n_in,
                              void* d_out, int out_size, void* d_ws, size_t ws_size,
                              hipStream_t stream) {
    const float* x   = (const float*)d_in[0];   // [N,128]
    const float* W1  = (const float*)d_in[1];   // [128,64]
    const float* b1  = (const float*)d_in[2];   // [64]
    const float* W2  = (const float*)d_in[3];   // [64,32]
    const float* b2  = (const float*)d_in[4];   // [32]
    const int*   pos = (const int*)d_in[5];     // [2,E]
    const int*   neg = (const int*)d_in[6];     // [2,E]
    float*       out = (float*)d_out;

    const int E = in_sizes[5] / 2;

    // workspace partition (floats): deg | dinv | xw[N,64] | h[N,64] | hw[N,32] | z[N,32]
    float* ws   = (float*)d_ws;
    float* deg  = ws;
    float* dinv = deg  + NNODES;
    float* xw   = dinv + NNODES;
    float* h    = xw   + (size_t)NNODES * 64;
    float* hw   = h    + (size_t)NNODES * 64;
    float* z    = hw   + (size_t)NNODES * 32;

    const int* prow = pos;       // source
    const int* pcol = pos + E;   // target

    auto blk = [](long long n) { return (unsigned)((n + THREADS - 1) / THREADS); };

    // normalization (shared by both conv layers: same graph)
    zero_f32   <<<blk(NNODES), THREADS, 0, stream>>>(deg, NNODES);
    count_indeg<<<blk(E),      THREADS, 0, stream>>>(pcol, E, deg);
    make_dinv  <<<blk(NNODES), THREADS, 0, stream>>>(deg, dinv, NNODES);

    // layer 1: xw = x@W1 ; h = Ahat @ xw ; h = relu(h + b1)
    const int mTiles = NNODES / 16;                         // 6250 exactly
    gemm_wmma_f32<<<blk((long long)mTiles * 4 * 32), THREADS, 0, stream>>>(
        x, W1, xw, mTiles, 128, 64);
    init_self   <<<blk((long long)NNODES * 64), THREADS, 0, stream>>>(
        xw, dinv, h, NNODES * 64, 6);
    scatter_edges<<<blk((long long)E * 64), THREADS, 0, stream>>>(
        prow, pcol, dinv, xw, h, E, 6);
    bias_act    <<<blk((long long)NNODES * 64), THREADS, 0, stream>>>(
        h, b1, NNODES * 64, 63, 1);

    // layer 2: hw = h@W2 ; z = Ahat @ hw ; z = z + b2
    gemm_wmma_f32<<<blk((long long)mTiles * 2 * 32), THREADS, 0, stream>>>(
        h, W2, hw, mTiles, 64, 32);
    init_self   <<<blk((long long)NNODES * 32), THREADS, 0, stream>>>(
        hw, dinv, z, NNODES * 32, 5);
    scatter_edges<<<blk((long long)E * 32), THREADS, 0, stream>>>(
        prow, pcol, dinv, hw, z, E, 5);
    bias_act    <<<blk((long long)NNODES * 32), THREADS, 0, stream>>>(
        z, b2, NNODES * 32, 31, 0);

    // decode over pos++neg edges; also emits edge_index into d_out tail
    decode<<<blk((long long)2 * E), THREADS, 0, stream>>>(pos, neg, z, out, E);
}